// UHGContrastiveLoss_21328807592549
// MI455X (gfx1250) — compile-verified
//
#include <hip/hip_runtime.h>
#include <hip/hip_bf16.h>

#define N_PTS   8192
#define DIM     64
#define NTILES  (N_PTS / 16)
#define NCLS    64
#define TEMP    0.07f
#define MARGINV 1.0f
#define BIGV    1e30f
#define LN2F    0.6931471805599453f
#define BSTR    68   // LDS row stride (floats); 68 % 64 == 4 -> conflict-free

typedef __attribute__((ext_vector_type(2))) float v2f;
typedef __attribute__((ext_vector_type(8))) float v8f;

struct Accum { float pos_sum; float neg_sum; float hard_sum; unsigned pos_cnt; unsigned neg_cnt; };

struct Top5 {
  float t0, t1, t2, t3, t4;
  __device__ void init() { t0 = t1 = t2 = t3 = t4 = BIGV; }
  // caller guarantees v < t4
  __device__ void push(float v) {
    t4 = v;
    float a, b;
    a = fminf(t3, t4); b = fmaxf(t3, t4); t3 = a; t4 = b;
    a = fminf(t2, t3); b = fmaxf(t2, t3); t2 = a; t3 = b;
    a = fminf(t1, t2); b = fmaxf(t1, t2); t1 = a; t2 = b;
    a = fminf(t0, t1); b = fmaxf(t0, t1); t0 = a; t1 = b;
  }
  __device__ void insert(float v) { if (v < t4) push(v); }
};

// Label histogram -> exact pair counts; also zero-initializes the accumulators.
__global__ void uhg_counts_init(const int* __restrict__ labels, float* ws) {
  __shared__ unsigned h[NCLS];
  int tid = threadIdx.x;
  if (tid < NCLS) h[tid] = 0u;
  __syncthreads();
  for (int i = tid; i < N_PTS; i += 256) atomicAdd(&h[labels[i]], 1u);
  __syncthreads();
  if (tid == 0) {
    unsigned ss = 0u;
    for (int c = 0; c < NCLS; ++c) ss += h[c] * h[c];
    Accum* a = (Accum*)(ws + N_PTS);
    a->pos_sum = 0.f; a->neg_sum = 0.f; a->hard_sum = 0.f;
    a->pos_cnt = ss - (unsigned)N_PTS;                               // sum n_c(n_c-1)
    a->neg_cnt = (unsigned)N_PTS * (N_PTS - 1) - (ss - (unsigned)N_PTS);
  }
}

// rs[i] = 1 / sqrt(|<e_i, e_i>_Minkowski|)   (EPS folded away; norms are O(D))
__global__ void uhg_row_norms(const float* __restrict__ emb, float* __restrict__ rs) {
  int i = blockIdx.x * blockDim.x + threadIdx.x;
  if (i >= N_PTS) return;
  const float4* r = (const float4*)(emb + i * DIM);
  float acc = 0.f;
#pragma unroll
  for (int k = 0; k < 16; ++k) {
    float4 v = r[k];
    acc += v.x * v.x + v.y * v.y + v.z * v.z + (k == 15 ? -v.w * v.w : v.w * v.w);
  }
  rs[i] = __builtin_amdgcn_rsqf(fabsf(acc));
}

__global__ __launch_bounds__(256)
void uhg_main(const float* __restrict__ emb, const int* __restrict__ labels,
              const float* __restrict__ rs, float* __restrict__ ws) {
  __shared__ float btile[2][16 * BSTR];     // double-buffered: 16 cols x 64 feats, pre-scaled
  __shared__ float merge[8 * 16 * 16 * 5];  // per wave: 16 rows x 16 lanes x top5

  const int tid  = threadIdx.x;
  const int wave = tid >> 5;
  const int lane = tid & 31;
  const int hi   = lane >> 4;       // half-wave select
  const int lc   = lane & 15;
  const int kbase = hi ? 2 : 0;     // f32 16x4 A/B layout: lanes 16-31 hold K=2,3

  const int r0   = blockIdx.x * 128 + wave * 16;
  const int rowA = r0 + lc;

  // Preload A fragments pre-scaled by 1/sqrt(|nrm_row|): WMMA then outputs G*rs_r*rs_c
  const float rsA = rs[rowA];
  v2f afrag[16];
#pragma unroll
  for (int k = 0; k < 16; ++k) {
    v2f a = *(const v2f*)(emb + rowA * DIM + 4 * k + kbase);
    a.x *= rsA; a.y *= rsA;
    afrag[k] = a;
  }

  // Per-lane row labels + diagonal-lane masks for the 8 accumulator slots (row = v + 8*hi)
  int  labR[8];
  bool diagLane[8];
#pragma unroll
  for (int v = 0; v < 8; ++v) {
    labR[v]     = labels[r0 + v + 8 * hi];
    diagLane[v] = (lc == v + 8 * hi);   // lane-constant: this lane's column hits row v+8hi
  }

  Top5 tp[8];
#pragma unroll
  for (int v = 0; v < 8; ++v) tp[v].init();

  float pos_sum = 0.f, neg_sum = 0.f;

  // B-tile staging (one float4 per thread), pre-scaled by rs[col], last feature sign-flipped
  const int   scol  = tid >> 4;                       // 0..15
  const int   sf4   = (tid & 15) * 4;                 // 0..60
  const float signw = ((tid & 15) == 15) ? -1.f : 1.f;

  float4 stage = *(const float4*)(emb + scol * DIM + sf4);
  {
    float c = rs[scol];
    stage.x *= c; stage.y *= c; stage.z *= c; stage.w *= c * signw;
  }
  int labC = labels[lc];   // column label for this lane, tile 0 (prestaged)

  for (int t = 0; t < NTILES; ++t) {
    const int buf = t & 1;
    *(float4*)(&btile[buf][scol * BSTR + sf4]) = stage;
    __syncthreads();

    const int labC_cur = labC;

    // Prefetch next tile + metadata into registers while WMMAs run
    if (t + 1 < NTILES) {
      stage = *(const float4*)(emb + ((t + 1) * 16 + scol) * DIM + sf4);
      float c = rs[(t + 1) * 16 + scol];
      stage.x *= c; stage.y *= c; stage.z *= c; stage.w *= c * signw;
      labC = labels[(t + 1) * 16 + lc];
    }

    // Two independent WMMA chains (even/odd K) to break the RAW accumulator chain
    v8f acc0 = {}, acc1 = {};
#pragma unroll
    for (int k = 0; k < 8; ++k) {
      v2f b0 = *(const v2f*)(&btile[buf][lc * BSTR + 8 * k + kbase]);
      v2f b1 = *(const v2f*)(&btile[buf][lc * BSTR + 8 * k + 4 + kbase]);
      acc0 = __builtin_amdgcn_wmma_f32_16x16x4_f32(false, afrag[2 * k],     false, b0,
                                                   (short)0, acc0, false, false);
      acc1 = __builtin_amdgcn_wmma_f32_16x16x4_f32(false, afrag[2 * k + 1], false, b1,
                                                   (short)0, acc1, false, false);
    }

    const bool diag_tile = (t * 16 == r0);   // uniform: only 1 of 512 tiles has the diagonal

#pragma unroll
    for (int v = 0; v < 8; ++v) {
      float ratio = fmaxf(fabsf(acc0[v] + acc1[v]), 1.0f + 1e-7f);  // |G|*rs_r*rs_c, clamped
      // arccosh(x) = ln2 * log2(x + sqrt(x^2 - 1)); arg >= 1+1e-7 -> raw fast paths safe
      float srt = __builtin_amdgcn_sqrtf(__builtin_fmaf(ratio, ratio, -1.0f));
      float d   = LN2F * __builtin_amdgcn_logf(ratio + srt);
      bool  same = (labC_cur == labR[v]);
      bool  diag = diag_tile && diagLane[v];
      pos_sum += (same && !diag) ? d : 0.f;   // branchless -> cndmask
      neg_sum += same ? 0.f : d;
      if (!same && d < tp[v].t4) tp[v].push(d);  // rare after warm-up
    }
    // single barrier per iteration: one barrier (iter t) always separates a buffer's
    // readers (iter t-1) from its next writer (iter t+1)
  }

  __syncthreads();

  // Merge per-lane top5 candidates across the 16 column lanes of each row
  float* mbase = &merge[wave * 16 * 16 * 5];
#pragma unroll
  for (int v = 0; v < 8; ++v) {
    int row = v + 8 * hi;
    float* p = &mbase[(row * 16 + lc) * 5];
    p[0] = tp[v].t0; p[1] = tp[v].t1; p[2] = tp[v].t2; p[3] = tp[v].t3; p[4] = tp[v].t4;
  }
  __syncthreads();

  float hard_local = 0.f;
  if (lane < 16) {                      // lane r merges row r of this wave's stripe
    Top5 m; m.init();
    const float* p = &mbase[lane * 16 * 5];
#pragma unroll 5
    for (int j = 0; j < 80; ++j) m.insert(p[j]);
    float tv[5] = { m.t0, m.t1, m.t2, m.t3, m.t4 };
    float cnt = 0.f, ssum = 0.f;
#pragma unroll
    for (int j = 0; j < 5; ++j) {
      if (tv[j] < BIGV * 0.5f) { cnt += 1.f; ssum += fmaxf(MARGINV - tv[j], 0.f); }
    }
    hard_local = ssum / fmaxf(cnt, 1.f);
  }

  // wave32 reductions
#pragma unroll
  for (int o = 16; o > 0; o >>= 1) {
    pos_sum    += __shfl_xor(pos_sum, o, 32);
    neg_sum    += __shfl_xor(neg_sum, o, 32);
    hard_local += __shfl_xor(hard_local, o, 32);
  }
  if (lane == 0) {
    Accum* a = (Accum*)(ws + N_PTS);
    atomicAdd(&a->pos_sum, pos_sum);
    atomicAdd(&a->neg_sum, neg_sum);
    atomicAdd(&a->hard_sum, hard_local);
  }
}

__global__ void uhg_finalize(const float* ws, float* out) {
  if (threadIdx.x != 0 || blockIdx.x != 0) return;
  const Accum* a = (const Accum*)(ws + N_PTS);
  float pc = (float)(a->pos_cnt > 1u ? a->pos_cnt : 1u);
  float nc = (float)(a->neg_cnt > 1u ? a->neg_cnt : 1u);
  float pos_loss = a->pos_sum / (TEMP * pc);   // -sum(sim_pos)/cnt = sum(d)/T/cnt
  float neg_loss = -a->neg_sum / (TEMP * nc);  //  sum(sim_neg)/cnt
  float contrastive = pos_loss + neg_loss;
  float hard = a->hard_sum / (float)N_PTS;
  out[0] = contrastive + hard;  // total
  out[1] = contrastive;
  out[2] = hard;
}

extern "C" void kernel_launch(void* const* d_in, const int* in_sizes, int n_in,
                              void* d_out, int out_size, void* d_ws, size_t ws_size,
                              hipStream_t stream) {
  (void)in_sizes; (void)n_in; (void)out_size; (void)ws_size;
  const float* emb    = (const float*)d_in[0];
  const int*   labels = (const int*)d_in[1];
  float* ws  = (float*)d_ws;
  float* out = (float*)d_out;

  hipLaunchKernelGGL(uhg_counts_init, dim3(1), dim3(256), 0, stream, labels, ws);
  hipLaunchKernelGGL(uhg_row_norms, dim3(N_PTS / 256), dim3(256), 0, stream, emb, ws);
  hipLaunchKernelGGL(uhg_main, dim3(N_PTS / 128), dim3(256), 0, stream, emb, labels, ws, ws);
  hipLaunchKernelGGL(uhg_finalize, dim3(1), dim3(32), 0, stream, ws, out);
}